// MambaBlock_65309272703270
// MI455X (gfx1250) — compile-verified
//
#include <hip/hip_runtime.h>
#include <hip/hip_bf16.h>
#include <math.h>

typedef __attribute__((ext_vector_type(2))) float v2f;
typedef __attribute__((ext_vector_type(8))) float v8f;
typedef __attribute__((ext_vector_type(4))) unsigned int u32x4;
typedef __attribute__((ext_vector_type(8))) int i32x8;
typedef __attribute__((ext_vector_type(4))) int i32x4;

#define D_MODEL  768
#define N_ST     16
#define D_INNER  1536            // 2*D_MODEL
#define DT_RANK  48
#define KSZ      4
#define BATCH    2
#define SEQ      1024
#define ROWS     (BATCH*SEQ)     // 2048
#define KB       32              // K-panel staged in LDS per step

// ---------------------------------------------------------------------------
// fp32 WMMA helpers (V_WMMA_F32_16X16X4_F32, wave32)
// A fragment (16x4): a[j] = A[r][k + 2*half + j]
// B fragment (4x16): b[j] = B[k + 2*half + j][n0 + r]
// C/D (16x16):       d[v] = D[v + 8*half][r]
// ---------------------------------------------------------------------------
__device__ __forceinline__ v8f v8zero() {
    v8f z = {0.f, 0.f, 0.f, 0.f, 0.f, 0.f, 0.f, 0.f};
    return z;
}

__device__ __forceinline__ void wmma_store(float* __restrict__ C, int ldc,
                                           int m0, int n0, const v8f& acc) {
    const int lane = threadIdx.x & 31;
    const int half = lane >> 4;
    const int r    = lane & 15;
#pragma unroll
    for (int v = 0; v < 8; ++v)
        C[(m0 + v + 8 * half) * ldc + n0 + r] = acc[v];
}

__device__ __forceinline__ float softplus_f(float v) {
    return v > 20.f ? v : log1pf(__expf(v));
}
__device__ __forceinline__ float swish_f(float v) {
    return v * (1.f / (1.f + __expf(-v)));
}

// Direct-from-global WMMA tile (used by the two small GEMMs)
template <int NT>
__device__ __forceinline__ void wmma_tile(const float* __restrict__ A,
                                          const float* __restrict__ Bm,
                                          int lda, int ldb, int K,
                                          int m0, int n0, v8f acc[NT]) {
    const int lane = threadIdx.x & 31;
    const int half = lane >> 4;
    const int r    = lane & 15;
    const float* arow = A + (m0 + r) * lda;
    for (int k = 0; k < K; k += 4) {
        v2f a;
        a.x = arow[k + 2 * half + 0];
        a.y = arow[k + 2 * half + 1];
#pragma unroll
        for (int t = 0; t < NT; ++t) {
            const float* bp = Bm + (k + 2 * half) * ldb + (n0 + t * 16 + r);
            v2f b;
            b.x = bp[0];
            b.y = bp[ldb];
            acc[t] = __builtin_amdgcn_wmma_f32_16x16x4_f32(
                false, a, false, b, (short)0, acc[t], false, false);
        }
    }
}

// ---------------------------------------------------------------------------
// Stage a KB x 64 fp32 B-panel into LDS (byte offset 0 of dynamic LDS) using
// the Tensor Data Mover. Descriptor per CDNA5 ISA.
// This toolchain exposes the 6-arg builtin:
//   (uint32x4 g0, int32x8 g1, int32x4 g2, int32x4 g3, int32x8 extra, i32 cpol)
// One TDM issue per workgroup (wave 0); caller must __syncthreads() after.
// ---------------------------------------------------------------------------
__device__ __forceinline__ void stage_B_panel(const float* __restrict__ gsrc,
                                              int ldb_elems, float* sB) {
#if __has_builtin(__builtin_amdgcn_tensor_load_to_lds)
    if (threadIdx.x < 32) {            // wave 0 only; TDM issues once per wave
        unsigned long long ga = (unsigned long long)(size_t)gsrc;
        const unsigned td0 = (unsigned)ldb_elems;   // tensor_dim0 (elements)
        const unsigned td1 = KB;                    // tensor_dim1 (rows)
        u32x4 g0;
        g0[0] = 1u;                                  // count=1, user descriptor
        g0[1] = 0u;                                  // lds_addr = 0 (dyn-LDS base)
        g0[2] = (unsigned)(ga & 0xffffffffu);        // global_addr[31:0]
        g0[3] = (unsigned)((ga >> 32) & 0x01ffffffu) // global_addr[56:32]
              | (2u << 30);                          // type = 2 ("image")
        i32x8 g1;
        g1[0] = (int)(2u << 16);                     // data_size = 4 bytes
        g1[1] = (int)((td0 & 0xffffu) << 16);        // tensor_dim0[15:0]
        g1[2] = (int)(((td0 >> 16) & 0xffffu) | ((td1 & 0xffffu) << 16));
        g1[3] = (int)(((td1 >> 16) & 0xffffu) | (64u << 16));  // tile_dim0 = 64
        g1[4] = (int)(KB & 0xffffu);                 // tile_dim1 = KB, tile_dim2 = 0
        g1[5] = (int)td0;                            // tensor_dim0_stride[31:0]
        g1[6] = 0;                                   // stride hi / dim1 stride lo
        g1[7] = 0;
        i32x4 gz4 = {0, 0, 0, 0};
        i32x8 gz8 = {0, 0, 0, 0, 0, 0, 0, 0};
        __builtin_amdgcn_tensor_load_to_lds(g0, g1, gz4, gz4, gz8, 0);
#if __has_builtin(__builtin_amdgcn_s_wait_tensorcnt)
        __builtin_amdgcn_s_wait_tensorcnt(0);
#else
        asm volatile("s_wait_tensorcnt 0x0" ::: "memory");
#endif
    }
#else
    // Fallback: cooperative copy through VGPRs
    for (int i = threadIdx.x; i < KB * 64; i += 256)
        sB[i] = gsrc[(i >> 6) * ldb_elems + (i & 63)];
#endif
}

// ---------------------------------------------------------------------------
// Large GEMM: C[ROWS x N] = A[ROWS x Kdim] @ B[Kdim x N]
// Block = 256 threads (8 waves) -> 128x64 C tile; B panel shared via LDS/TDM.
// ---------------------------------------------------------------------------
template <int Kdim, int LDA, int LDB, int LDC, int NBLKS>
__global__ void k_gemm_big(const float* __restrict__ A,
                           const float* __restrict__ Bm,
                           float* __restrict__ C) {
    extern __shared__ float sB[];                    // [KB][64] floats
    const int mblk = blockIdx.x / NBLKS;
    const int nblk = blockIdx.x % NBLKS;
    const int wid  = threadIdx.x >> 5;
    const int lane = threadIdx.x & 31;
    const int half = lane >> 4;
    const int r    = lane & 15;
    const int m0 = mblk * 128 + wid * 16;
    const int n0 = nblk * 64;

    v8f acc[4] = {v8zero(), v8zero(), v8zero(), v8zero()};
    const float* arow = A + (m0 + r) * LDA;

    for (int k0 = 0; k0 < Kdim; k0 += KB) {
        stage_B_panel(Bm + (size_t)k0 * LDB + n0, LDB, sB);
        __syncthreads();
        __builtin_prefetch(arow + k0 + KB, 0, 1);    // global_prefetch_b8
#pragma unroll
        for (int kk = 0; kk < KB; kk += 4) {
            v2f a;
            a.x = arow[k0 + kk + 2 * half + 0];
            a.y = arow[k0 + kk + 2 * half + 1];
#pragma unroll
            for (int t = 0; t < 4; ++t) {
                const float* bp = &sB[(kk + 2 * half) * 64 + t * 16 + r];
                v2f b;
                b.x = bp[0];
                b.y = bp[64];
                acc[t] = __builtin_amdgcn_wmma_f32_16x16x4_f32(
                    false, a, false, b, (short)0, acc[t], false, false);
            }
        }
        __syncthreads();
    }
#pragma unroll
    for (int t = 0; t < 4; ++t)
        wmma_store(C, LDC, m0, n0 + t * 16, acc[t]);
}

// ---------------------------------------------------------------------------
// depthwise causal conv (K=4) + bias + swish on xi = xr[:, :, :D_INNER]
// ---------------------------------------------------------------------------
__global__ void k_conv_swish(const float* __restrict__ xr,
                             const float* __restrict__ conv_w,
                             const float* __restrict__ conv_b,
                             float* __restrict__ xi) {
    const int idx = blockIdx.x * blockDim.x + threadIdx.x;   // < ROWS*D_INNER
    const int d   = idx % D_INNER;
    const int row = idx / D_INNER;                           // b*SEQ + l
    const int l   = row % SEQ;
    const int rowbase = row * (2 * D_INNER) + d;
    float acc = conv_b[d];
#pragma unroll
    for (int k = 0; k < KSZ; ++k) {
        int lk = l - (KSZ - 1) + k;
        if (lk >= 0)
            acc += conv_w[d * KSZ + k] * xr[rowbase + (lk - l) * (2 * D_INNER)];
    }
    xi[idx] = swish_f(acc);
}

// ---------------------------------------------------------------------------
// x_dbl = xi @ W_x   (2048x1536 @ 1536x80 -> 2048x80), NT=5, 128 waves
// ---------------------------------------------------------------------------
__global__ void k_gemm_xdbl(const float* __restrict__ xi,
                            const float* __restrict__ W_x,
                            float* __restrict__ xdbl) {
    const int wave = (blockIdx.x * blockDim.x + threadIdx.x) >> 5;  // 0..127
    const int m0 = wave * 16;
    v8f acc[5] = {v8zero(), v8zero(), v8zero(), v8zero(), v8zero()};
    wmma_tile<5>(xi, W_x, D_INNER, DT_RANK + 2 * N_ST, D_INNER, m0, 0, acc);
#pragma unroll
    for (int t = 0; t < 5; ++t)
        wmma_store(xdbl, DT_RANK + 2 * N_ST, m0, t * 16, acc[t]);
}

// ---------------------------------------------------------------------------
// delta = softplus(x_dbl[:, :48] @ W_dt + b_dt)   (2048x48 @ 48x1536)
// ---------------------------------------------------------------------------
__global__ void k_gemm_delta(const float* __restrict__ xdbl,
                             const float* __restrict__ W_dt,
                             const float* __restrict__ b_dt,
                             float* __restrict__ delta) {
    const int wave = (blockIdx.x * blockDim.x + threadIdx.x) >> 5;
    const int mt  = wave / 24;        // 0..127
    const int nt4 = wave % 24;        // 0..23
    const int m0 = mt * 16, n0 = nt4 * 64;
    v8f acc[4] = {v8zero(), v8zero(), v8zero(), v8zero()};
    wmma_tile<4>(xdbl, W_dt, DT_RANK + 2 * N_ST, D_INNER, DT_RANK, m0, n0, acc);
    const int lane = threadIdx.x & 31;
    const int half = lane >> 4;
    const int r    = lane & 15;
#pragma unroll
    for (int t = 0; t < 4; ++t) {
        const int col = n0 + t * 16 + r;
        const float bias = b_dt[col];
#pragma unroll
        for (int v = 0; v < 8; ++v)
            delta[(m0 + v + 8 * half) * D_INNER + col] = softplus_f(acc[t][v] + bias);
    }
}

// ---------------------------------------------------------------------------
// SSM scan + skip (u*Dp) + swish(res) gating
// one lane per (b, d); 16 states in VGPRs; B/C staged through LDS
// ---------------------------------------------------------------------------
__global__ void k_scan(const float* __restrict__ delta,
                       const float* __restrict__ xi,
                       const float* __restrict__ xdbl,
                       const float* __restrict__ xr,
                       const float* __restrict__ A_log,
                       const float* __restrict__ Dp,
                       float* __restrict__ ys) {
    const int b = blockIdx.x / (D_INNER / 256);
    const int d = (blockIdx.x % (D_INNER / 256)) * 256 + threadIdx.x;
    __shared__ float sB[N_ST];
    __shared__ float sC[N_ST];

    float Arow[N_ST], h[N_ST];
#pragma unroll
    for (int n = 0; n < N_ST; ++n) {
        Arow[n] = -__expf(A_log[d * N_ST + n]);
        h[n] = 0.f;
    }
    const float dp = Dp[d];

    for (int l = 0; l < SEQ; ++l) {
        const int row = b * SEQ + l;
        const int t = threadIdx.x;
        if (t < N_ST)
            sB[t] = xdbl[row * (DT_RANK + 2 * N_ST) + DT_RANK + t];
        else if (t < 2 * N_ST)
            sC[t - N_ST] = xdbl[row * (DT_RANK + 2 * N_ST) + DT_RANK + N_ST + (t - N_ST)];
        __syncthreads();

        const float dlt = delta[row * D_INNER + d];
        const float u   = xi[row * D_INNER + d];
        const float du  = dlt * u;
        float y = 0.f;
#pragma unroll
        for (int n = 0; n < N_ST; ++n) {
            const float w = __expf(dlt * Arow[n]);
            h[n] = w * h[n] + du * sB[n];
            y += h[n] * sC[n];
        }
        y += u * dp;
        const float res = xr[row * (2 * D_INNER) + D_INNER + d];
        ys[row * D_INNER + d] = y * swish_f(res);
        __syncthreads();
    }
}

// ---------------------------------------------------------------------------
extern "C" void kernel_launch(void* const* d_in, const int* in_sizes, int n_in,
                              void* d_out, int out_size, void* d_ws, size_t ws_size,
                              hipStream_t stream) {
    const float* x      = (const float*)d_in[0];
    const float* W_in   = (const float*)d_in[1];
    const float* conv_w = (const float*)d_in[2];
    const float* conv_b = (const float*)d_in[3];
    const float* W_x    = (const float*)d_in[4];
    const float* W_dt   = (const float*)d_in[5];
    const float* b_dt   = (const float*)d_in[6];
    const float* A_log  = (const float*)d_in[7];
    const float* Dp     = (const float*)d_in[8];
    const float* W_out  = (const float*)d_in[9];
    float* out = (float*)d_out;

    float* xr    = (float*)d_ws;                       // ROWS * 2*D_INNER
    float* xi    = xr    + (size_t)ROWS * 2 * D_INNER; // ROWS * D_INNER
    float* xdbl  = xi    + (size_t)ROWS * D_INNER;     // ROWS * 80
    float* delta = xdbl  + (size_t)ROWS * (DT_RANK + 2 * N_ST); // ROWS * D_INNER
    float* ys    = delta + (size_t)ROWS * D_INNER;     // ROWS * D_INNER

    const size_t lds_bytes = KB * 64 * sizeof(float);  // 8 KB B panel

    // 1) in-projection: 16 mblks x 48 nblks = 768 blocks, TDM-staged B panels
    k_gemm_big<D_MODEL, D_MODEL, 2 * D_INNER, 2 * D_INNER, 48>
        <<<768, 256, lds_bytes, stream>>>(x, W_in, xr);
    // 2) conv + swish
    k_conv_swish<<<(ROWS * D_INNER) / 256, 256, 0, stream>>>(xr, conv_w, conv_b, xi);
    // 3) x_dbl: 128 waves -> 16 blocks x 256
    k_gemm_xdbl<<<16, 256, 0, stream>>>(xi, W_x, xdbl);
    // 4) delta: 3072 waves -> 384 blocks x 256
    k_gemm_delta<<<384, 256, 0, stream>>>(xdbl, W_dt, b_dt, delta);
    // 5) scan: BATCH * D_INNER lanes -> 12 blocks x 256
    k_scan<<<BATCH * (D_INNER / 256), 256, 0, stream>>>(delta, xi, xdbl, xr, A_log, Dp, ys);
    // 6) out-projection: 16 mblks x 12 nblks = 192 blocks, TDM-staged B panels
    k_gemm_big<D_INNER, D_INNER, D_MODEL, D_MODEL, 12>
        <<<192, 256, lds_bytes, stream>>>(ys, W_out, out);
}